// QuantumLayer_37666863185963
// MI455X (gfx1250) — compile-verified
//
#include <hip/hip_runtime.h>

// ---------------------------------------------------------------------------
// QNN expectation values on MI455X (gfx1250, wave32).
//
// Math:  state after the RY feature map is REAL:
//          v_n = prod_i (bit_i(n) ? sin : cos)(x_i/2).
//        <Z_w> = v^T M_w v,  M_w = sum_k z_w(k) (Ur_k Ur_k^T + Ui_k Ui_k^T),
//        U = 16x16 unitary of the variational layers (weights only).
//
// Plan:  kernel1 builds the four M_w pre-packed in WMMA *A* layout -> d_ws.
//        kernel2, per wave / 16-sample tile, all in registers (no LDS memory):
//          stage1: W_w = M_w @ V^T  (A = packed M_w, B = a[] regs; the
//                  B-layout of V^T is bit-identical to V's A-layout regs)
//          close:  ev_w[m] = sum_idx V[m][idx] * W_w[idx][m].  Lane (m,h)
//                  holds W rows [8h,8h+8) in its C-layout regs; a single
//                  tile-wide cross-half V exchange (4x shfl_xor(16)) gives it
//                  the matching V components, so the diagonal costs 8 FMAs +
//                  one shfl_xor + add per observable.
//        => 16 v_wmma_f32_16x16x4_f32 + 8 ds_bpermute per tile, zero LDS mem.
// Bandwidth-bound (~24MB traffic @ 23.3TB/s ~= 1us); f32 WMMA keeps precision.
// ---------------------------------------------------------------------------

typedef __attribute__((ext_vector_type(2))) float v2f;
typedef __attribute__((ext_vector_type(8))) float v8f;

// ---------------- Kernel 1: build packed M matrices (tiny) ------------------
__global__ __launch_bounds__(256) void qnn_build_M(const float* __restrict__ wts,
                                                   float* __restrict__ Mout) {
  __shared__ float Ur[16][16];
  __shared__ float Ui[16][16];
  const int t = threadIdx.x;

  if (t < 16) {
    // column t of U: apply variational circuit to basis state e_t.
    float ar[16], ai[16];
#pragma unroll
    for (int k = 0; k < 16; ++k) { ar[k] = (k == t) ? 1.f : 0.f; ai[k] = 0.f; }

#pragma unroll
    for (int l = 0; l < 2; ++l) {
#pragma unroll
      for (int i = 0; i < 4; ++i) {
        const int mask = 8 >> i;  // wire i <-> bit (3-i)
        // RX(theta): [[c, -i s],[-i s, c]]
        {
          const float th = wts[l * 8 + i * 2 + 0];
          const float c = __cosf(0.5f * th), s = __sinf(0.5f * th);
#pragma unroll
          for (int n = 0; n < 16; ++n) {
            if (!(n & mask)) {
              const int n1 = n | mask;
              const float a0r = ar[n], a0i = ai[n], a1r = ar[n1], a1i = ai[n1];
              ar[n]  = c * a0r + s * a1i;  ai[n]  = c * a0i - s * a1r;
              ar[n1] = c * a1r + s * a0i;  ai[n1] = c * a1i - s * a0r;
            }
          }
        }
        // RZ(theta): diag(e^{-i th/2}, e^{+i th/2})
        {
          const float th = wts[l * 8 + i * 2 + 1];
          const float c = __cosf(0.5f * th), s = __sinf(0.5f * th);
#pragma unroll
          for (int n = 0; n < 16; ++n) {
            if (!(n & mask)) {
              const int n1 = n | mask;
              const float a0r = ar[n], a0i = ai[n], a1r = ar[n1], a1i = ai[n1];
              ar[n]  = c * a0r + s * a0i;  ai[n]  = c * a0i - s * a0r;
              ar[n1] = c * a1r - s * a1i;  ai[n1] = c * a1i + s * a1r;
            }
          }
        }
      }
      // CNOT ring: (0,1),(1,2),(2,3),(3,0)
#pragma unroll
      for (int e = 0; e < 4; ++e) {
        const int ctrl = (e < 3) ? e : 3;
        const int tgt  = (e < 3) ? e + 1 : 0;
        const int cm = 8 >> ctrl, tm = 8 >> tgt;
#pragma unroll
        for (int n = 0; n < 16; ++n) {
          if ((n & cm) && !(n & tm)) {
            const int n1 = n | tm;
            const float tr = ar[n], ti = ai[n];
            ar[n] = ar[n1];  ai[n] = ai[n1];
            ar[n1] = tr;     ai[n1] = ti;
          }
        }
      }
    }
#pragma unroll
    for (int k = 0; k < 16; ++k) { Ur[k][t] = ar[k]; Ui[k][t] = ai[k]; }
  }
  __syncthreads();

  // Pack M_w into WMMA *A* layout (lane holds row M = lane%16):
  //   Mout[((w*4+kc)*32 + lane)*2 + r] = M_w[lane%16][4kc + r + (lane<16?0:2)]
  // (M_w is symmetric, so row/col roles are interchangeable.)
#pragma unroll
  for (int q = 0; q < 4; ++q) {
    const int e = t + q * 256;
    const int r = e & 1;
    const int lane = (e >> 1) & 31;
    const int kc = (e >> 6) & 3;
    const int w = (e >> 8) & 3;
    const int row = lane & 15;
    const int col = kc * 4 + r + ((lane < 16) ? 0 : 2);
    float sum = 0.f;
#pragma unroll
    for (int k = 0; k < 16; ++k) {
      const float sgn = ((k >> (3 - w)) & 1) ? -1.f : 1.f;
      sum += sgn * (Ur[k][row] * Ur[k][col] + Ui[k][row] * Ui[k][col]);
    }
    Mout[e] = sum;
  }
}

// ---------------- Kernel 2: batched quadratic forms via WMMA ----------------
__global__ __launch_bounds__(256) void qnn_eval(const float* __restrict__ x,
                                                const float* __restrict__ M,
                                                float* __restrict__ out,
                                                int T, int total_waves) {
  const int lane = threadIdx.x & 31;
  const int m    = lane & 15;   // sample within tile (N index)
  const int half = lane >> 4;   // 0: lanes 0-15, 1: lanes 16-31
  const int wave_id = blockIdx.x * 8 + (threadIdx.x >> 5);

  // Packed M_w chunks in A-operand registers (held for whole kernel).
  v2f Am[4][4];
#pragma unroll
  for (int w = 0; w < 4; ++w) {
#pragma unroll
    for (int kc = 0; kc < 4; ++kc) {
      const float* p = M + ((w * 4 + kc) * 32 + lane) * 2;
      v2f b; b.x = p[0]; b.y = p[1];
      Am[w][kc] = b;
    }
  }

  for (int t = wave_id; t < T; t += total_waves) {
    const int s0 = t * 16;

    // Prefetch next grid-stride tile's x row (global_prefetch_b8); clamp to
    // keep the speculative address inside the buffer.
    {
      int tn = t + total_waves;
      tn = (tn < T) ? tn : t;
      __builtin_prefetch(reinterpret_cast<const float4*>(x) + tn * 16 + m, 0, 1);
    }

    // Amplitudes of this lane's sample: v_idx = p01[idx>>2] * p23[idx&3]
    const float4 xv = reinterpret_cast<const float4*>(x)[s0 + m];
    float sn0, cs0, sn1, cs1, sn2, cs2, sn3, cs3;
    __sincosf(0.5f * xv.x, &sn0, &cs0);
    __sincosf(0.5f * xv.y, &sn1, &cs1);
    __sincosf(0.5f * xv.z, &sn2, &cs2);
    __sincosf(0.5f * xv.w, &sn3, &cs3);
    const float p01[4] = {cs0 * cs1, cs0 * sn1, sn0 * cs1, sn0 * sn1};
    const float q0 = half ? (sn2 * cs3) : (cs2 * cs3);  // idx&3 = 2*half + r
    const float q1 = half ? (sn2 * sn3) : (cs2 * sn3);

    // a[kc] = V components idx = 4kc + 2*half + r  (A-layout of V and,
    // identically, B-layout of V^T).
    v2f a[4];
#pragma unroll
    for (int kc = 0; kc < 4; ++kc) {
      v2f av; av.x = p01[kc] * q0; av.y = p01[kc] * q1;
      a[kc] = av;
    }

    // Tile-wide cross-half V exchange (observable-independent, done once):
    // each lane sends the 4 components its partner needs.
    //   h=0 sends V[8],V[9],V[12],V[13] (a[2],a[3]); receives V[2,3,6,7].
    //   h=1 sends V[2],V[3],V[6],V[7]  (a[0],a[1]); receives V[8,9,12,13].
    const float sd0 = half ? a[0].x : a[2].x;
    const float sd1 = half ? a[0].y : a[2].y;
    const float sd2 = half ? a[1].x : a[3].x;
    const float sd3 = half ? a[1].y : a[3].y;
    const float rc0 = __shfl_xor(sd0, 16, 32);
    const float rc1 = __shfl_xor(sd1, 16, 32);
    const float rc2 = __shfl_xor(sd2, 16, 32);
    const float rc3 = __shfl_xor(sd3, 16, 32);
    // Vrow[r] = V[m][r + 8*half]  -- matches this lane's W rows.
    float Vrow[8];
    Vrow[0] = half ? rc0    : a[0].x;
    Vrow[1] = half ? rc1    : a[0].y;
    Vrow[2] = half ? a[2].x : rc0;
    Vrow[3] = half ? a[2].y : rc1;
    Vrow[4] = half ? rc2    : a[1].x;
    Vrow[5] = half ? rc3    : a[1].y;
    Vrow[6] = half ? a[3].x : rc2;
    Vrow[7] = half ? a[3].y : rc3;

    float ow[4];
#pragma unroll
    for (int w = 0; w < 4; ++w) {
      // Stage 1: W_w = M_w @ V^T  (C-layout: lane holds W_w[r+8h][m])
      v8f Wc = {};
#pragma unroll
      for (int kc = 0; kc < 4; ++kc) {
        Wc = __builtin_amdgcn_wmma_f32_16x16x4_f32(
            false, Am[w][kc], false, a[kc], (short)0, Wc, false, false);
      }
      // Half of the diagonal dot: sum over this lane's 8 W rows.
      float part = Wc[0] * Vrow[0];
#pragma unroll
      for (int r = 1; r < 8; ++r) part = fmaf(Wc[r], Vrow[r], part);
      // Other half lives in the cross-half partner (same m).
      ow[w] = part + __shfl_xor(part, 16, 32);
    }

    // Lanes 0-15 now hold ev[m] for all 4 observables: coalesced float4 store.
    if (half == 0) {
      float4 o;
      o.x = ow[0]; o.y = ow[1]; o.z = ow[2]; o.w = ow[3];
      reinterpret_cast<float4*>(out)[s0 + m] = o;
    }
  }
}

// ---------------------------------------------------------------------------
extern "C" void kernel_launch(void* const* d_in, const int* in_sizes, int n_in,
                              void* d_out, int out_size, void* d_ws, size_t ws_size,
                              hipStream_t stream) {
  const float* x   = (const float*)d_in[0];   // [B,4] f32
  const float* wts = (const float*)d_in[1];   // [2,4,2] f32
  float* out = (float*)d_out;                 // [B,4] f32
  float* Mws = (float*)d_ws;                  // 1024 floats (4KB)

  const int B = in_sizes[0] / 4;
  const int T = B / 16;  // 16-sample tiles (B = 524288 -> T = 32768)

  qnn_build_M<<<1, 256, 0, stream>>>(wts, Mws);

  const int blocks = 512;                 // 4096 waves, grid-stride over tiles
  const int total_waves = blocks * 8;
  qnn_eval<<<blocks, 256, 0, stream>>>(x, Mws, out, T, total_waves);
}